// FusionAndOutput_54640573939741
// MI455X (gfx1250) — compile-verified
//
#include <hip/hip_runtime.h>

// fp32 WMMA fragment types for V_WMMA_F32_16X16X4_F32 (wave32):
//   A: 16x4 f32  -> 2 VGPRs/lane ; B: 4x16 f32 -> 2 VGPRs/lane ; C/D: 16x16 f32 -> 8 VGPRs/lane
typedef float v2f __attribute__((ext_vector_type(2)));
typedef float v8f __attribute__((ext_vector_type(8)));

__device__ __forceinline__ v8f wmma_f32(v2f a, v2f b, v8f c) {
  // 8 args: (neg_a, A, neg_b, B, c_mod, C, reuse_a, reuse_b)
  return __builtin_amdgcn_wmma_f32_16x16x4_f32(false, a, false, b,
                                               (short)0, c, false, false);
}

// B=16, T=24, N=512, D=128, Z=12. Rows = B*N = 8192, K1 = 3*D = 384.
// Grid: 512 blocks (one 16-row tile each), 256 threads (8 waves).
__global__ __launch_bounds__(256)
void fusion_head_wmma(const float* __restrict__ Ht,
                      const float* __restrict__ Hs,
                      const float* __restrict__ Hm,
                      const float* __restrict__ W1,   // [128, 384] row-major
                      const float* __restrict__ b1,   // [128]
                      const float* __restrict__ W2,   // [12, 128] row-major
                      const float* __restrict__ b2,   // [12]
                      float* __restrict__ out)        // [16, 12, 512]
{
  __shared__ float ytile[16 * 128];  // post-activation Y tile, 8 KB

  const int tile = blockIdx.x;        // 0..511
  const int wave = threadIdx.x >> 5;  // 0..7  -> channel block
  const int lane = threadIdx.x & 31;
  const int b    = tile >> 5;         // 32 tiles per batch (512/16)
  const int n0   = (tile & 31) << 4;  // first n of this tile

  const int m  = lane & 15;           // row (A/M) or column (B/N) index
  const int hi = lane >> 4;           // half-wave selector (K sub-pair)

  // Last timestep base: ((b*T + (T-1))*N + n0)*D
  const long base = ((long)(b * 24 + 23) * 512 + n0) * 128;
  const float* srcA[3] = { Ht + base, Hs + base, Hm + base };

  const int ch = wave * 16 + m;                 // output channel d (B-matrix N)
  const float* w1row = W1 + (long)ch * 384;

  // ---------------- GEMM1: Y = X(16x384) * W1^T(384x128), this wave: 16 cols
  v8f acc = {};
  #pragma unroll
  for (int s = 0; s < 3; ++s) {
    const float* ap = srcA[s] + m * 128 + 2 * hi;   // A: X[row m][c0 + 2*hi .. +1]
    const float* bp = w1row + s * 128 + 2 * hi;     // B: W1[ch][c0 + 2*hi .. +1]
    #pragma unroll
    for (int kk = 0; kk < 32; ++kk) {
      v2f afrag = *(const v2f*)(ap + kk * 4);
      v2f bfrag = *(const v2f*)(bp + kk * 4);
      acc = wmma_f32(afrag, bfrag, acc);
    }
  }

  // Bias + LeakyReLU in C-layout (VGPR i: row = i + 8*hi, col = ch), stage to LDS
  const float bias1 = b1[ch];
  #pragma unroll
  for (int i = 0; i < 8; ++i) {
    float v = acc[i] + bias1;
    v = (v > 0.0f) ? v : 0.01f * v;
    ytile[(i + 8 * hi) * 128 + ch] = v;
  }

  __syncthreads();

  // ---------------- GEMM2: Z = Y(16x128) * W2^T(128x16, 12 valid) on wave 0
  if (wave == 0) {
    v8f acc2 = {};
    const bool zv = (m < 12);                 // B-matrix column z = m
    const float* w2row = W2 + m * 128;
    #pragma unroll
    for (int k = 0; k < 32; ++k) {
      const int c0 = 4 * k + 2 * hi;
      v2f afrag = *(const v2f*)(&ytile[m * 128 + c0]);   // A: Y[m][c0 .. c0+1]
      v2f bfrag = {};
      if (zv) bfrag = *(const v2f*)(w2row + c0);         // B: W2[z][c0 .. c0+1]
      acc2 = wmma_f32(afrag, bfrag, acc2);
    }
    if (zv) {
      const float bias2 = b2[m];
      // out[b][z][n0 + row], row = i + 8*hi per D-matrix layout
      float* orow = out + ((long)b * 12 + m) * 512 + n0;
      #pragma unroll
      for (int i = 0; i < 8; ++i)
        orow[i + 8 * hi] = acc2[i] + bias2;
    }
  }
}

extern "C" void kernel_launch(void* const* d_in, const int* in_sizes, int n_in,
                              void* d_out, int out_size, void* d_ws, size_t ws_size,
                              hipStream_t stream) {
  (void)in_sizes; (void)n_in; (void)out_size; (void)d_ws; (void)ws_size;
  const float* Ht = (const float*)d_in[0];
  const float* Hs = (const float*)d_in[1];
  const float* Hm = (const float*)d_in[2];
  const float* W1 = (const float*)d_in[3];
  const float* b1 = (const float*)d_in[4];
  const float* W2 = (const float*)d_in[5];
  const float* b2 = (const float*)d_in[6];
  float* out = (float*)d_out;

  dim3 grid(512), block(256);
  hipLaunchKernelGGL(fusion_head_wmma, grid, block, 0, stream,
                     Ht, Hs, Hm, W1, b1, W2, b2, out);
}